// QNet_7825430413374
// MI455X (gfx1250) — compile-verified
//
#include <hip/hip_runtime.h>

typedef float v2f __attribute__((ext_vector_type(2)));
typedef float v8f __attribute__((ext_vector_type(8)));
typedef int   v4i __attribute__((vector_size(16)));   // matches builtin param type

#define EMB    128
#define NNODES 256
#define NBATCH 16

// ---- CDNA5 async global->LDS copy helpers ----
#if __has_builtin(__builtin_amdgcn_global_load_async_to_lds_b128)
#define HAVE_ASYNC_LDS 1
#else
#define HAVE_ASYNC_LDS 0
#endif

typedef __attribute__((address_space(1))) v4i* gv4i_p;
typedef __attribute__((address_space(3))) v4i* lv4i_p;

__device__ __forceinline__ void async_cp16(const float* g, float* l) {
#if HAVE_ASYNC_LDS
    __builtin_amdgcn_global_load_async_to_lds_b128(
        (gv4i_p)(float*)g, (lv4i_p)l, 0, 0);
#else
    *(float4*)l = *(const float4*)g;
#endif
}

__device__ __forceinline__ void wait_async() {
#if HAVE_ASYNC_LDS
#if __has_builtin(__builtin_amdgcn_s_wait_asynccnt)
    __builtin_amdgcn_s_wait_asynccnt(0);
#else
    asm volatile("s_wait_asynccnt 0" ::: "memory");
#endif
#endif
}

// h1 = relu(xv @ t1_w + t1_b); K=5, VALU is the right tool.
__global__ void k_theta1(const float* __restrict__ xv, const float* __restrict__ t1w,
                         const float* __restrict__ t1b, float* __restrict__ h1) {
    int idx = blockIdx.x * blockDim.x + threadIdx.x;   // 4096*128 threads
    int e = idx & (EMB - 1);
    int r = idx >> 7;                                   // b*N + n
    const float* x = xv + r * 5;
    float acc = t1b[e];
#pragma unroll
    for (int d = 0; d < 5; ++d) acc += x[d] * t1w[d * EMB + e];
    h1[idx] = fmaxf(acc, 0.f);
}

// s3sum[b,j,e] = sum_i relu(Ws[b,i,j]*t4w[e] + t4b[e])
__global__ void k_s3sum(const float* __restrict__ Ws, const float* __restrict__ t4w,
                        const float* __restrict__ t4b, float* __restrict__ s3s) {
    int bj = blockIdx.x;                 // b*N + j
    int e  = threadIdx.x;                // 0..127
    int b  = bj >> 8, j = bj & (NNODES - 1);
    const float* wcol = Ws + (long)b * NNODES * NNODES + j;
    float we = t4w[e], be = t4b[e];
    float acc = 0.f;
    for (int i = 0; i < NNODES; ++i)
        acc += fmaxf(wcol[i * NNODES] * we + be, 0.f);
    s3s[(long)bj * EMB + e] = acc;
}

// fp32 WMMA GEMM, 8 waves/block covering one 16-row strip of D (128 cols).
// A tile (16xK, lda==K) staged once per block into LDS via async copy; fragments
// read with ds_load_2addr_b64 (row stride padded to K+4 -> conflict-free banks).
// ldb == ldd == 128 are compile-time so B loads / D stores use immediate offsets.
// A-fragment layout (16x4 f32, ISA 7.12.2): lane<16 -> K={0,1}, lane>=16 -> K={2,3}.
template <int K, bool CONN, bool RELU, bool BIAS, bool ADD>
__global__ void k_gemm8(const float* __restrict__ A, const float* __restrict__ Bm,
                        const float* __restrict__ bias, const float* __restrict__ add,
                        float* __restrict__ D,
                        long sA, long sB, long sD, int mtiles) {
    constexpr int LD = EMB;                   // ldb == ldd == 128
    constexpr int KP = K + 4;                 // padded LDS row stride (floats)
    __shared__ float As[16 * KP];

    int blk   = blockIdx.x;
    int batch = blk / mtiles;
    int mt    = blk - batch * mtiles;
    int tid   = threadIdx.x;

    // Stage the 16xK A tile (async, 16B per thread per issue, lda == K so the
    // tile is one contiguous 16*K-float span of A).
    const float* Ag = A + (long)batch * sA + (long)mt * (16 * K);
    constexpr int CH = 16 * K / 4;            // float4 chunks
#pragma unroll
    for (int q = tid; q < CH; q += 256) {
        int row = q >> (K == 128 ? 5 : 6);    // q / (K/4)
        int c   = (q - row * (K / 4)) * 4;
        async_cp16(Ag + row * K + c, &As[row * KP + c]);
    }
    wait_async();
    __syncthreads();

    int wave = tid >> 5, lane = tid & 31;
    int half = lane >> 4, lc = lane & 15;
    int nt   = wave;                           // 8 waves -> 8 N-tiles
    int col  = nt * 16 + lc;

    const float* Bb = Bm + (long)batch * sB + 2 * half * LD + col;
    const float* Al = &As[lc * KP + 2 * half];

    v8f acc = {};
#pragma unroll 4
    for (int kk = 0; kk < K; kk += 4) {
        v2f a = *(const v2f*)(Al + kk);        // ds_load_b64 (pairs -> 2addr)
        if (CONN) {
            a.x = a.x > 0.f ? 1.f : 0.f;
            a.y = a.y > 0.f ? 1.f : 0.f;
        }
        const float* bp = Bb + kk * LD;
        v2f b;
        b.x = bp[0];
        b.y = bp[LD];
        acc = __builtin_amdgcn_wmma_f32_16x16x4_f32(false, a, false, b,
                                                    (short)0, acc, false, false);
    }

    // Epilogue: single base pointer, immediate row offsets (v*LD floats).
    float bv = BIAS ? bias[col] : 0.f;
    long base = (long)batch * sD + (long)(mt * 16 + 8 * half) * LD + col;
    float*       Dp = D + base;
    const float* Ap = ADD ? add + base : nullptr;
#pragma unroll
    for (int v = 0; v < 8; ++v) {
        float x = acc[v] + bv;
        if (ADD) x += Ap[v * LD];
        if (RELU) x = fmaxf(x, 0.f);
        Dp[v * LD] = x;
    }
}

__global__ void k_zero(float* __restrict__ p, int n) {
    int i = blockIdx.x * blockDim.x + threadIdx.x;
    if (i < n) p[i] = 0.f;
}

// musum[b,e] = sum_n mu[b,n,e]
__global__ void k_musum(const float* __restrict__ mu, float* __restrict__ msum) {
    int b = blockIdx.x, e = threadIdx.x;
    const float* p = mu + (long)b * NNODES * EMB + e;
    float acc = 0.f;
    for (int n = 0; n < NNODES; ++n) acc += p[n * EMB];
    msum[b * EMB + e] = acc;
}

// out[b,n] = t5b + sum_e gsr[b,e]*t5w[e] + lrl[b,n,e]*t5w[128+e]
__global__ void k_out(const float* __restrict__ gsr, const float* __restrict__ lrl,
                      const float* __restrict__ t5w, const float* __restrict__ t5b,
                      float* __restrict__ out) {
    int bn = blockIdx.x;                 // 4096
    int e  = threadIdx.x;                // 128
    int b  = bn >> 8;
    float v = gsr[b * EMB + e] * t5w[e] + lrl[(long)bn * EMB + e] * t5w[EMB + e];
    __shared__ float red[128];
    red[e] = v;
    __syncthreads();
    for (int s = 64; s > 0; s >>= 1) {
        if (e < s) red[e] += red[e + s];
        __syncthreads();
    }
    if (e == 0) out[bn] = red[0] + t5b[0];
}

extern "C" void kernel_launch(void* const* d_in, const int* in_sizes, int n_in,
                              void* d_out, int out_size, void* d_ws, size_t ws_size,
                              hipStream_t stream) {
    (void)in_sizes; (void)n_in; (void)out_size; (void)ws_size;
    const float* xv   = (const float*)d_in[0];
    const float* Ws   = (const float*)d_in[1];
    const float* t1w  = (const float*)d_in[2];
    const float* t1b  = (const float*)d_in[3];
    const float* t2w  = (const float*)d_in[4];
    const float* t2b  = (const float*)d_in[5];
    const float* t3w  = (const float*)d_in[6];
    const float* t3b  = (const float*)d_in[7];
    const float* t4w  = (const float*)d_in[8];
    const float* t4b  = (const float*)d_in[9];
    const float* t5w  = (const float*)d_in[10];
    const float* t5b  = (const float*)d_in[11];
    const float* t6w  = (const float*)d_in[12];
    const float* t6b  = (const float*)d_in[13];
    const float* t7w  = (const float*)d_in[14];
    const float* t7b  = (const float*)d_in[15];
    const float* layw = (const float*)d_in[16];
    const float* layb = (const float*)d_in[17];
    float* out = (float*)d_out;
    float* ws  = (float*)d_ws;

    const long S = (long)NBATCH * NNODES * EMB;   // 524288 floats
    float* h1   = ws;
    float* s13  = ws + S;
    float* s3s  = ws + 2 * S;
    float* mu   = ws + 3 * S;
    float* tmp  = ws + 4 * S;
    float* lrl  = ws + 5 * S;
    float* msum = ws + 6 * S;            // 16*128
    float* gsr  = ws + 6 * S + 2048;     // 16*128

    // h1 = relu(theta1(xv))
    k_theta1<<<2048, 256, 0, stream>>>(xv, t1w, t1b, h1);
    // s13 = h1 @ layer_w + layer_b          (4096x128x128)
    k_gemm8<128, false, false, true, false><<<256, 256, 0, stream>>>(
        h1, layw, layb, nullptr, s13, 0, 0, 0, 256);
    // s3sum
    k_s3sum<<<4096, 128, 0, stream>>>(Ws, t4w, t4b, s3s);
    // s13 += s3sum @ t3_w + t3_b
    k_gemm8<128, false, false, true, true><<<256, 256, 0, stream>>>(
        s3s, t3w, t3b, s13, s13, 0, 0, 0, 256);
    // mu = 0
    k_zero<<<2048, 256, 0, stream>>>(mu, (int)S);

    for (int t = 0; t < 4; ++t) {
        // tmp[b] = conn[b] @ mu[b]   (batched 256x128x256, conn from Ws>0 on the fly)
        k_gemm8<256, true, false, false, false><<<256, 256, 0, stream>>>(
            Ws, mu, nullptr, nullptr, tmp,
            (long)NNODES * NNODES, (long)NNODES * EMB, (long)NNODES * EMB, 16);
        // mu = relu(s13 + tmp @ t2_w + t2_b)
        k_gemm8<128, false, true, true, true><<<256, 256, 0, stream>>>(
            tmp, t2w, t2b, s13, mu, 0, 0, 0, 256);
    }

    // pooled state
    k_musum<<<16, 128, 0, stream>>>(mu, msum);
    // gsr = relu(msum @ t6_w + t6_b)   (16x128x128, one block)
    k_gemm8<128, false, true, true, false><<<1, 256, 0, stream>>>(
        msum, t6w, t6b, nullptr, gsr, 0, 0, 0, 1);
    // lrl = relu(mu @ t7_w + t7_b)
    k_gemm8<128, false, true, true, false><<<256, 256, 0, stream>>>(
        mu, t7w, t7b, nullptr, lrl, 0, 0, 0, 256);
    // final dot with t5
    k_out<<<4096, 128, 0, stream>>>(gsr, lrl, t5w, t5b, out);
}